// Conv3d_77799037600002
// MI455X (gfx1250) — compile-verified
//
#include <hip/hip_runtime.h>

typedef __attribute__((ext_vector_type(2))) float v2f;
typedef __attribute__((ext_vector_type(8))) float v8f;

#define CIN  32
#define COUT 64
#define LDS_PITCH 65   // pad: avoids 2-way bank conflict between lane halves

__global__ void bias_init_kernel(float* __restrict__ out,
                                 const float* __restrict__ bias,
                                 int total, int cout) {
    int i = blockIdx.x * blockDim.x + threadIdx.x;
    if (i < total) out[i] = bias[i % cout];
}

__global__ __launch_bounds__(256) void sparse_conv_wmma_kernel(
    const float* __restrict__ feat,   // [N, CIN]
    const int*   __restrict__ kmap,   // [K, M, 2] (in_idx, out_idx)
    const float* __restrict__ kw,     // [K, CIN, COUT]
    float*       __restrict__ out,    // [N, COUT], pre-filled with bias
    int M, int tiles_per_offset)
{
    __shared__ float sw[CIN * LDS_PITCH];     // 8320 B weight tile, padded pitch

    const int k_off = blockIdx.y;             // kernel offset (grid.y = K)
    const int tid   = threadIdx.x;

    // ---- Stage this offset's 32x64 weight tile into LDS (coalesced) ----
    const float* kwk = kw + (size_t)k_off * CIN * COUT;
#pragma unroll
    for (int i = tid; i < CIN * COUT; i += 256) {
        sw[(i >> 6) * LDS_PITCH + (i & 63)] = kwk[i];
    }
    __syncthreads();

    // ---- One wave = one 16-row tile of this offset ----
    const int tile = blockIdx.x * (blockDim.x >> 5) + (tid >> 5);
    if (tile >= tiles_per_offset) return;     // wave-uniform (after barrier)

    const int lane = tid & 31;
    const int r16  = lane & 15;               // A-row within half / B,C column
    const int kh   = lane >> 4;               // K-half select
    const int row_base = tile * 16;

    const int* kmap_k = kmap + (size_t)k_off * M * 2;

    // ---- Gather A tile: 16 rows x CIN, striped per ISA 16x4 f32 A layout ----
    int arow   = row_base + r16;
    int arow_c = arow < M ? arow : (M - 1);
    int in_idx = kmap_k[(size_t)arow_c * 2 + 0];

    const float* frow = feat + (size_t)in_idx * CIN;
    v2f a[8];
#pragma unroll
    for (int kb = 0; kb < 8; ++kb) {
        int col = kb * 4 + kh * 2;            // {K, K+1} contiguous, 8B aligned
        a[kb] = *(const v2f*)(frow + col);
    }

    // ---- Output scatter indices for the C rows this lane will store ----
    int  oidx[8];
    bool oval[8];
#pragma unroll
    for (int j = 0; j < 8; ++j) {
        int crow = row_base + kh * 8 + j;     // C layout: VGPR j -> M = j + 8*kh
        oval[j]  = crow < M;
        int cc   = oval[j] ? crow : (M - 1);
        oidx[j]  = kmap_k[(size_t)cc * 2 + 1];
    }

#pragma unroll
    for (int nt = 0; nt < 4; ++nt) {          // COUT = 4 x 16-wide tiles
        const int col = nt * 16 + r16;
        v8f acc = {};
#pragma unroll
        for (int kb = 0; kb < 8; ++kb) {      // CIN = 8 x K=4 steps
            int krow = kb * 4 + kh * 2;       // B 4x16 layout mirrors A
            v2f b;
            b.x = sw[krow * LDS_PITCH + col];
            b.y = sw[(krow + 1) * LDS_PITCH + col];
            acc = __builtin_amdgcn_wmma_f32_16x16x4_f32(
                false, a[kb], false, b, (short)0, acc, false, false);
        }
#pragma unroll
        for (int j = 0; j < 8; ++j) {
            if (oval[j]) {
                float* dst = out + (size_t)oidx[j] * COUT + col;
                __hip_atomic_fetch_add(dst, acc[j], __ATOMIC_RELAXED,
                                       __HIP_MEMORY_SCOPE_AGENT);
            }
        }
    }
}

extern "C" void kernel_launch(void* const* d_in, const int* in_sizes, int n_in,
                              void* d_out, int out_size, void* d_ws, size_t ws_size,
                              hipStream_t stream) {
    const float* feat = (const float*)d_in[0];
    // d_in[1] = input_coord (unused), d_in[2] = input_cmap (unused)
    const int*   kmap = (const int*)d_in[3];
    const float* kw   = (const float*)d_in[4];
    const float* bias = (const float*)d_in[5];
    float* out = (float*)d_out;

    const int K = in_sizes[4] / (CIN * COUT);   // 27
    const int M = in_sizes[3] / (K * 2);        // 100000

    // out = bias (broadcast), then accumulate scatter contributions.
    int total = out_size;
    bias_init_kernel<<<(total + 255) / 256, 256, 0, stream>>>(out, bias, total, COUT);

    const int tiles_per_offset = (M + 15) / 16; // 6250
    const int waves_per_block  = 8;             // 256 threads = 8 wave32
    dim3 grid((tiles_per_offset + waves_per_block - 1) / waves_per_block, K);
    sparse_conv_wmma_kernel<<<grid, 256, 0, stream>>>(
        feat, kmap, kw, out, M, tiles_per_offset);
}